// LightGCN_68410239091164
// MI455X (gfx1250) — compile-verified
//
#include <hip/hip_runtime.h>

// LightGCN forward, pull-based (CSR) implementation for gfx1250 (wave32).
//   deg[c]   = #edges with col==c
//   dinv[i]  = deg>0 ? 1/sqrt(deg) : 0
//   conv(h)[n] = dinv[n] * sum_{e: col[e]==n} dinv[row[e]] * h[row[e]]
//   out = (x + conv(x) + conv(conv(x))) / 3
//
// No fp atomics in the hot loops: CSR is built once per call (int atomics),
// then each node's 64-float embedding lives in one 16-lane half-wave
// (float4 per lane), accumulated in registers, stored with B128.
// Per-edge weights dinv[row] are materialized during bucketing so both conv
// passes stream (index, weight) coalesced; the only random access left is
// the unavoidable 256B x[row] gather (L2-resident: 38.4 MB << 192 MB L2).

#define EMB_DIM   64
#define DV        (EMB_DIM / 4)   // float4 per node row
#define SCAN_T    256
#define SCAN_ELEMS 1024           // elements scanned per block (256 thr * 4)

// ---- CDNA5 async global->LDS copy (guarded; falls back to plain loads) ----
#if defined(__has_builtin)
#  if __has_builtin(__builtin_amdgcn_global_load_async_to_lds_b128)
#    define HAVE_ASYNC_LDS 1
#  else
#    define HAVE_ASYNC_LDS 0
#  endif
#else
#  define HAVE_ASYNC_LDS 0
#endif

#if HAVE_ASYNC_LDS
// B128 payload type: builtin expects pointers to a 4-int vector
// (AS1 global source, AS3 LDS destination).
typedef int v4i __attribute__((__vector_size__(16)));
typedef __attribute__((address_space(1))) v4i* v4i_gptr;
typedef __attribute__((address_space(3))) v4i* v4i_lptr;

__device__ __forceinline__ void async_wait0() {
#if defined(__has_builtin) && __has_builtin(__builtin_amdgcn_s_wait_asynccnt)
    __builtin_amdgcn_s_wait_asynccnt(0);
#else
    asm volatile("s_wait_asynccnt 0x0" ::: "memory");
#endif
}
#endif

// ---------------- small utility kernels ----------------

__global__ void k_zero_i32(int* __restrict__ p, int n) {
    int i = blockIdx.x * blockDim.x + threadIdx.x;
    if (i < n) p[i] = 0;
}

__global__ void k_degree(const int* __restrict__ col, int E, int* __restrict__ degi) {
    int i = blockIdx.x * blockDim.x + threadIdx.x;
    if (i < E) atomicAdd(&degi[col[i]], 1);
}

__global__ void k_dinv(const int* __restrict__ degi, float* __restrict__ dinv, int N) {
    int i = blockIdx.x * blockDim.x + threadIdx.x;
    if (i < N) {
        int d = degi[i];
        dinv[i] = (d > 0) ? (1.0f / sqrtf((float)d)) : 0.0f;
    }
}

// ---------------- exclusive prefix scan over degrees (3 phases) ----------------

// Phase A: each block scans SCAN_ELEMS degrees -> block-local exclusive scan
// into colptr, block total into partial[blockIdx]. Input tile is staged to
// LDS with CDNA5 async copies when available (each thread reads back only
// its own B128, so only s_wait_asynccnt is needed, no barrier).
__global__ __launch_bounds__(SCAN_T)
void k_scan_block(const int* __restrict__ degi, int N,
                  int* __restrict__ colptr, int* __restrict__ partial) {
    __shared__ int s[SCAN_T];
    const int t    = threadIdx.x;
    const int base = blockIdx.x * SCAN_ELEMS + t * 4;

    int v[4];
    int sum = 0;
#if HAVE_ASYNC_LDS
    __shared__ __align__(16) int tile[SCAN_ELEMS];
    const bool full = (base + 4 <= N);
    if (full) {
        __builtin_amdgcn_global_load_async_to_lds_b128(
            (v4i_gptr)(degi + base),          // global source (AS1, v4i*)
            (v4i_lptr)(&tile[t * 4]),         // LDS destination (AS3, v4i*)
            /*imm offset*/0, /*cpol*/0);
    }
    async_wait0();
#pragma unroll
    for (int j = 0; j < 4; ++j) {
        int g = base + j;
        v[j] = full ? tile[t * 4 + j] : ((g < N) ? degi[g] : 0);
        sum += v[j];
    }
#else
#pragma unroll
    for (int j = 0; j < 4; ++j) {
        int g = base + j;
        v[j] = (g < N) ? degi[g] : 0;
        sum += v[j];
    }
#endif
    s[t] = sum;
    __syncthreads();
    // inclusive Hillis-Steele scan over per-thread sums
    for (int off = 1; off < SCAN_T; off <<= 1) {
        int x = (t >= off) ? s[t - off] : 0;
        __syncthreads();
        s[t] += x;
        __syncthreads();
    }
    int run = (t == 0) ? 0 : s[t - 1];
    if (t == SCAN_T - 1) partial[blockIdx.x] = s[SCAN_T - 1];
#pragma unroll
    for (int j = 0; j < 4; ++j) {
        int g = base + j;
        if (g < N) colptr[g] = run;
        run += v[j];
    }
}

// Phase B: exclusive-scan the (<=256) block totals in a single block.
__global__ __launch_bounds__(SCAN_T)
void k_scan_partials(int* __restrict__ partial, int nchunks) {
    __shared__ int s[SCAN_T];
    const int t = threadIdx.x;
    int v = (t < nchunks) ? partial[t] : 0;
    s[t] = v;
    __syncthreads();
    for (int off = 1; off < SCAN_T; off <<= 1) {
        int x = (t >= off) ? s[t - off] : 0;
        __syncthreads();
        s[t] += x;
        __syncthreads();
    }
    int excl = (t == 0) ? 0 : s[t - 1];
    if (t < nchunks) partial[t] = excl;
}

// Phase C: add chunk offsets; also initialize bucket cursors and colptr[N].
__global__ void k_scan_fixup(int* __restrict__ colptr, const int* __restrict__ partial,
                             int N, int E, int* __restrict__ cursor) {
    int i = blockIdx.x * blockDim.x + threadIdx.x;
    if (i < N) {
        int v = colptr[i] + partial[i / SCAN_ELEMS];
        colptr[i] = v;
        cursor[i] = v;
    }
    if (i == 0) colptr[N] = E;
}

// Bucket the edges by target node and materialize per-edge source weights:
// rowidx[colptr[c] .. colptr[c+1]) = source rows, roww[..] = dinv[source].
__global__ void k_place(const int* __restrict__ row, const int* __restrict__ col, int E,
                        const float* __restrict__ dinv,
                        int* __restrict__ cursor,
                        int* __restrict__ rowidx, float* __restrict__ roww) {
    int i = blockIdx.x * blockDim.x + threadIdx.x;
    if (i < E) {
        int r   = row[i];
        int c   = col[i];
        int pos = atomicAdd(&cursor[c], 1);
        rowidx[pos] = r;
        roww[pos]   = dinv[r];
    }
}

// ---------------- pull-based LGConv ----------------
// One node per 16-lane half-wave; each lane owns one float4 of the embedding.
// hout[n] = scale * ( dinv[n]*sum_r dinv[r]*hin[r]  (+ addA[n] + addB[n]) )
// Neighbor batches of 16 (index, weight) pairs are streamed coalesced and
// broadcast to the half-wave via __shfl width=16 (ds_bpermute).
__global__ __launch_bounds__(256)
void k_conv(const float4* __restrict__ hin,
            const int*    __restrict__ colptr,
            const int*    __restrict__ rowidx,
            const float*  __restrict__ roww,
            const float*  __restrict__ dinv,
            int N,
            float4*       __restrict__ hout,
            const float4* __restrict__ addA,   // nullable
            const float4* __restrict__ addB,   // nullable
            float scale) {
    int tid  = blockIdx.x * blockDim.x + threadIdx.x;
    int lane = tid & (DV - 1);
    int n    = tid >> 4;
    if (n >= N) return;

    const int beg = colptr[n];
    const int end = colptr[n + 1];

    float4 acc = make_float4(0.f, 0.f, 0.f, 0.f);
    for (int kb = beg; kb < end; kb += 16) {
        __builtin_prefetch(&rowidx[kb + 16], 0, 1);  // global_prefetch on CSR stream
        __builtin_prefetch(&roww[kb + 16],   0, 1);
        const int m = (end - kb < 16) ? (end - kb) : 16;
        int   myr = 0;
        float myw = 0.0f;
        if (lane < m) {
            myr = rowidx[kb + lane];   // coalesced 64B load per half-wave
            myw = roww[kb + lane];     // coalesced (no random dinv gather)
        }
        for (int j = 0; j < m; ++j) {
            int   r = __shfl(myr, j, 16);            // ds_bpermute broadcast
            float w = __shfl(myw, j, 16);
            float4 v = hin[(size_t)r * DV + lane];   // global_load_b128 gather
            acc.x = fmaf(v.x, w, acc.x);
            acc.y = fmaf(v.y, w, acc.y);
            acc.z = fmaf(v.z, w, acc.z);
            acc.w = fmaf(v.w, w, acc.w);
        }
    }
    float dn = dinv[n];
    acc.x *= dn; acc.y *= dn; acc.z *= dn; acc.w *= dn;

    if (addA != nullptr) {
        size_t o = (size_t)n * DV + lane;
        float4 a = addA[o];
        float4 b = addB[o];
        acc.x += a.x + b.x;
        acc.y += a.y + b.y;
        acc.z += a.z + b.z;
        acc.w += a.w + b.w;
    }
    acc.x *= scale; acc.y *= scale; acc.z *= scale; acc.w *= scale;
    hout[(size_t)n * DV + lane] = acc;
}

// ---------------- host launcher ----------------

extern "C" void kernel_launch(void* const* d_in, const int* in_sizes, int n_in,
                              void* d_out, int out_size, void* d_ws, size_t ws_size,
                              hipStream_t stream) {
    const float* x  = (const float*)d_in[0];
    const int*   ei = (const int*)d_in[1];
    const int N = in_sizes[0] / EMB_DIM;   // 150000
    const int E = in_sizes[1] / 2;         // 2000000
    const int* row = ei;                   // edge_index[0]
    const int* col = ei + E;               // edge_index[1]
    float* out = (float*)d_out;

    // Workspace carve-out (~56.8 MB total, 256B-aligned slices).
    char*  w = (char*)d_ws;
    size_t o = 0;
    auto carve = [&](size_t bytes) -> void* {
        void* p = w + o;
        o += (bytes + 255) & ~(size_t)255;
        return p;
    };
    float* h1      = (float*)carve((size_t)N * EMB_DIM * sizeof(float)); // 38.4 MB
    float* dinv    = (float*)carve((size_t)N * sizeof(float));
    int*   degi    = (int*)  carve((size_t)N * sizeof(int));
    int*   colptr  = (int*)  carve((size_t)(N + 1) * sizeof(int));
    int*   cursor  = (int*)  carve((size_t)N * sizeof(int));
    int*   partial = (int*)  carve((size_t)SCAN_T * sizeof(int));
    int*   rowidx  = (int*)  carve((size_t)E * sizeof(int));             // 8 MB
    float* roww    = (float*)carve((size_t)E * sizeof(float));           // 8 MB
    (void)ws_size; (void)n_in; (void)out_size;

    const int T = 256;
    const int nchunks = (N + SCAN_ELEMS - 1) / SCAN_ELEMS;               // 147 <= 256

    // 1) degree histogram (only buffer that needs per-call zeroing)
    k_zero_i32<<<(N + T - 1) / T, T, 0, stream>>>(degi, N);
    k_degree  <<<(E + T - 1) / T, T, 0, stream>>>(col, E, degi);
    k_dinv    <<<(N + T - 1) / T, T, 0, stream>>>(degi, dinv, N);

    // 2) CSR offsets via 3-phase exclusive scan + edge bucketing (with weights)
    k_scan_block   <<<nchunks, SCAN_T, 0, stream>>>(degi, N, colptr, partial);
    k_scan_partials<<<1,       SCAN_T, 0, stream>>>(partial, nchunks);
    k_scan_fixup   <<<(N + T - 1) / T, T, 0, stream>>>(colptr, partial, N, E, cursor);
    k_place        <<<(E + T - 1) / T, T, 0, stream>>>(row, col, E, dinv, cursor,
                                                       rowidx, roww);

    // 3) two pull-based LGConv passes; second pass fuses the 3-layer mean
    const long long work = (long long)N * DV;  // 16 lanes per node
    const int convBlocks = (int)((work + T - 1) / T);
    k_conv<<<convBlocks, T, 0, stream>>>((const float4*)x, colptr, rowidx, roww, dinv, N,
                                         (float4*)h1, nullptr, nullptr, 1.0f);
    k_conv<<<convBlocks, T, 0, stream>>>((const float4*)h1, colptr, rowidx, roww, dinv, N,
                                         (float4*)out, (const float4*)x, (const float4*)h1,
                                         1.0f / 3.0f);
}